// EEM_3573412790704
// MI455X (gfx1250) — compile-verified
//
#include <hip/hip_runtime.h>
#include <hip/hip_bf16.h>
#include <math.h>

// EEM atomic charges on MI455X (gfx1250, wave32).
// One workgroup (256 threads = 8 waves) per molecule. Whole 128x128 system
// lives in LDS. Border constraint eliminated via Schur complement:
//   A z1 = -Xi, A z2 = 1, lambda = (sum z1 - Qtot)/(sum z2), q = z1 - lambda*z2.
// Blocked unpivoted LU (NB=16); trailing update uses V_WMMA_F32_16X16X4_F32.

#define NATOMS 128
#define LDA    133          // odd stride -> conflict-free column walks on 64 banks
#define NB     16
#define NPANEL (NATOMS / NB)

typedef float v2f __attribute__((ext_vector_type(2)));
typedef float v8f __attribute__((ext_vector_type(8)));

__global__ void __launch_bounds__(256)
eem_solve_kernel(const int*   __restrict__ Z,      // [B*N]
                 const float* __restrict__ R,      // [B*N,3]
                 const float* __restrict__ Xia,    // [B*N]
                 const float* __restrict__ Qtot,   // [B]
                 const float* __restrict__ alpha,  // [NZ]
                 const float* __restrict__ Jtab,   // [NZ]
                 float*       __restrict__ out)    // [B*N]
{
    extern __shared__ float sA[];                  // [NATOMS * LDA] fp32 matrix

    __shared__ float sx[NATOMS], sy[NATOMS], sz[NATOMS];
    __shared__ float sAl2[NATOMS], sDiag[NATOMS];
    __shared__ float srhs0[NATOMS], srhs1[NATOMS];
    __shared__ float sPiv;
    __shared__ float sLam;

    const int tid  = threadIdx.x;
    const int mol  = blockIdx.x;
    const int base = mol * NATOMS;

    // ---- Stage 1: gather per-atom tables -------------------------------
    if (tid < NATOMS) {
        const int   g  = base + tid;
        const int   zi = Z[g];
        const float al = alpha[zi];
        sx[tid]   = R[3 * g + 0];
        sy[tid]   = R[3 * g + 1];
        sz[tid]   = R[3 * g + 2];
        sAl2[tid] = al * al;
        // Ad = J + (2/sqrt(pi)) * 1/(sqrt(2)*alpha)
        sDiag[tid] = Jtab[zi] + 1.1283791670955126f * (0.7071067811865476f / al);
        srhs0[tid] = -Xia[g];   // rhs for z1
        srhs1[tid] = 1.0f;      // rhs for z2
    }
    __syncthreads();

    // ---- Stage 2: build A (erf-screened Coulomb) in LDS ----------------
    for (int idx = tid; idx < NATOMS * NATOMS; idx += 256) {
        const int r = idx >> 7;
        const int c = idx & (NATOMS - 1);
        float val;
        if (r == c) {
            val = sDiag[r];
        } else {
            const float dx = sx[r] - sx[c];
            const float dy = sy[r] - sy[c];
            const float dz = sz[r] - sz[c];
            const float d2 = fmaxf(dx * dx + dy * dy + dz * dz, 0.0f);
            const float rij = sqrtf(d2) + 1e-20f;
            const float g   = rsqrtf(sAl2[r] + sAl2[c]);
            val = (rij > 1e-14f) ? (erff(g * rij) / rij) : 0.0f;
        }
        sA[r * LDA + c] = val;
    }
    __syncthreads();

    // ---- Stage 3: blocked LU, NB=16, WMMA fp32 trailing update ---------
    const int wid  = tid >> 5;
    const int lane = tid & 31;
    const int lm   = lane & 15;          // N (or M for A-frag) within tile
    const int mo   = (lane >> 4) << 3;   // C/D row offset: 0 or 8
    const int khl  = (lane >> 4) << 1;   // A/B K offset: 0 or 2

    for (int kb = 0; kb < NPANEL; ++kb) {
        const int k0 = kb * NB;
        const int k1 = k0 + NB;

        // 3a. panel factorization: columns k0..k0+15, rows k0..127
        for (int j = 0; j < NB; ++j) {
            const int jj = k0 + j;
            if (tid == 0) sPiv = 1.0f / sA[jj * LDA + jj];
            __syncthreads();
            {   // scale L column below pivot
                const int r = jj + 1 + tid;
                if (r < NATOMS) sA[r * LDA + jj] *= sPiv;
            }
            __syncthreads();
            {   // rank-1 update of the remaining panel columns only
                const int nrow  = NATOMS - (jj + 1);
                const int ncol  = (k1 - 1) - jj;
                const int total = nrow * ncol;
                for (int t = tid; t < total; t += 256) {
                    const int rr = jj + 1 + t / ncol;
                    const int cc = jj + 1 + t % ncol;
                    sA[rr * LDA + cc] =
                        fmaf(-sA[rr * LDA + jj], sA[jj * LDA + cc], sA[rr * LDA + cc]);
                }
            }
            __syncthreads();
        }

        const int ntc = NATOMS - k1;     // trailing size (multiple of 16)
        if (ntc > 0) {
            // 3b. U12 = L11^-1 * A12 (unit-lower triangular solve, 16 rows)
            if (tid < ntc) {
                const int c = k1 + tid;
                for (int j = 1; j < NB; ++j) {
                    const int jj = k0 + j;
                    float s = sA[jj * LDA + c];
                    for (int i = 0; i < j; ++i)
                        s = fmaf(-sA[jj * LDA + (k0 + i)], sA[(k0 + i) * LDA + c], s);
                    sA[jj * LDA + c] = s;
                }
            }
            __syncthreads();

            // 3c. trailing GEMM: A22 -= L21 * U12 via V_WMMA_F32_16X16X4_F32
            const int T = ntc >> 4;      // tiles per dimension
            for (int tile = wid; tile < T * T; tile += 8) {   // wave-uniform
                const int ti   = tile / T;
                const int tj   = tile - ti * T;
                const int row0 = k1 + ti * 16;
                const int col0 = k1 + tj * 16;

                // load C tile (ISA 7.12.2 C/D layout)
                v8f acc;
#pragma unroll
                for (int r = 0; r < 8; ++r)
                    acc[r] = sA[(row0 + mo + r) * LDA + col0 + lm];

                // K=16 as 4 chained K=4 WMMAs; negate A since fp32 WMMA
                // has no A/B NEG (only C-negate).
#pragma unroll
                for (int q = 0; q < 4; ++q) {
                    const int ak = k0 + q * 4 + khl;
                    v2f a, b;
                    a[0] = -sA[(row0 + lm) * LDA + ak];
                    a[1] = -sA[(row0 + lm) * LDA + ak + 1];
                    b[0] = sA[ak * LDA + col0 + lm];
                    b[1] = sA[(ak + 1) * LDA + col0 + lm];
                    acc = __builtin_amdgcn_wmma_f32_16x16x4_f32(
                        false, a, false, b, (short)0, acc, false, false);
                }

#pragma unroll
                for (int r = 0; r < 8; ++r)
                    sA[(row0 + mo + r) * LDA + col0 + lm] = acc[r];
            }
            __syncthreads();
        }
    }

    // ---- Stage 4: forward substitution (L unit-lower), 2 RHS -----------
    for (int j = 0; j < NATOMS - 1; ++j) {
        __syncthreads();
        const float y0 = srhs0[j];
        const float y1 = srhs1[j];
        if (tid > j && tid < NATOMS) {
            const float l = sA[tid * LDA + j];
            srhs0[tid] = fmaf(-l, y0, srhs0[tid]);
            srhs1[tid] = fmaf(-l, y1, srhs1[tid]);
        }
    }

    // ---- Stage 5: back substitution (U), 2 RHS -------------------------
    for (int j = NATOMS - 1; j >= 0; --j) {
        __syncthreads();
        if (tid == 0) {
            const float inv = 1.0f / sA[j * LDA + j];
            srhs0[j] *= inv;
            srhs1[j] *= inv;
        }
        __syncthreads();
        const float x0 = srhs0[j];
        const float x1 = srhs1[j];
        if (tid < j) {
            const float u = sA[tid * LDA + j];
            srhs0[tid] = fmaf(-u, x0, srhs0[tid]);
            srhs1[tid] = fmaf(-u, x1, srhs1[tid]);
        }
    }
    __syncthreads();

    // ---- Stage 6: Schur combine + store --------------------------------
    if (tid == 0) {
        float s1 = 0.0f, s2 = 0.0f;
        for (int i = 0; i < NATOMS; ++i) { s1 += srhs0[i]; s2 += srhs1[i]; }
        sLam = (s1 - Qtot[mol]) / s2;
    }
    __syncthreads();
    if (tid < NATOMS) {
        out[base + tid] = fmaf(-sLam, srhs1[tid], srhs0[tid]);
    }
}

extern "C" void kernel_launch(void* const* d_in, const int* in_sizes, int n_in,
                              void* d_out, int out_size, void* d_ws, size_t ws_size,
                              hipStream_t stream) {
    (void)n_in; (void)d_ws; (void)ws_size; (void)out_size;
    const int*   Z     = (const int*)  d_in[0];
    const float* R     = (const float*)d_in[1];
    const float* Xia   = (const float*)d_in[2];
    const float* Qtot  = (const float*)d_in[3];
    const float* alpha = (const float*)d_in[4];
    const float* Jtab  = (const float*)d_in[5];
    float*       out   = (float*)d_out;

    const int B = in_sizes[3];                       // Q_tot has one entry per molecule
    const size_t smem = (size_t)NATOMS * LDA * sizeof(float);  // ~68 KB (< 320 KB/WGP)

    eem_solve_kernel<<<dim3(B), dim3(256), smem, stream>>>(
        Z, R, Xia, Qtot, alpha, Jtab, out);
}